// AudioPCWrapper_70970039599632
// MI455X (gfx1250) — compile-verified
//
#include <hip/hip_runtime.h>

// AudioPCWrapper: CIRM decompression + 4-dir complex Gram-Schmidt.
// Memory-bound point-wise op (131.6 MB traffic -> ~5.6us @ 23.3 TB/s).
// No matmul structure -> no WMMA; wave32 VALU streaming kernel with b128 IO,
// v_log_f32/v_rsq_f32 transcendentals, NT stores, one global_prefetch_b8.
// 2D grid (y = batch) removes all integer division from addressing.

typedef float v4f __attribute__((ext_vector_type(4)));

namespace {
constexpr float LIMIT = 9.9f;
constexpr float EPS   = 1e-8f;
constexpr float Kc    = 10.0f;
constexpr float KLN2  = 6.93147180559945309417e+0f; // K * ln(2)
constexpr int   FT    = 257 * 1000;                 // F*T plane size in floats
constexpr int   NCH   = 8;                          // 2 * n_dirs channels
constexpr int   GPB   = FT / 4;                     // 64250 float4 groups / plane

// -K*ln((K-m)/(K+m)) = K*ln2 * (log2(K+m) - log2(K-m)); args in [0.1,19.9] > 0.
__device__ __forceinline__ float decompress(float m) {
    m = fminf(fmaxf(m, -LIMIT), LIMIT);
    return KLN2 * (__builtin_amdgcn_logf(Kc + m) - __builtin_amdgcn_logf(Kc - m));
}
} // namespace

__global__ __launch_bounds__(256) void gs_crm_kernel(
        const float* __restrict__ in, float* __restrict__ out) {
    const int x = blockIdx.x * blockDim.x + threadIdx.x; // float4 group in plane
    if (x >= GPB) return;
    const long base = (long)blockIdx.y * NCH * FT + ((long)x << 2);

    // Warm the L2 read stream ahead of the dependent b128 burst (gfx1250 path).
    __builtin_prefetch(in + base, 0, 3);

    __align__(16) float ch[NCH][4];
    #pragma unroll
    for (int c = 0; c < NCH; ++c)
        *(v4f*)ch[c] = *(const v4f*)(in + base + (long)c * FT); // global_load_b128

    __align__(16) float o[NCH][4];
    #pragma unroll
    for (int k = 0; k < 4; ++k) {
        float vr[4], vi[4];
        #pragma unroll
        for (int d = 0; d < 4; ++d) {
            vr[d] = decompress(ch[2 * d    ][k]);
            vi[d] = decompress(ch[2 * d + 1][k]);
        }
        // basis 0: v0 / |v0 + EPS|   (EPS added to the real part)
        float t   = vr[0] + EPS;
        float inv = __builtin_amdgcn_rsqf(t * t + vi[0] * vi[0]);
        float br  = vr[0] * inv, bi = vi[0] * inv;
        o[0][k] = br; o[1][k] = bi;
        float Sr = br, Si = bi; // S = sum of basis vectors so far
        #pragma unroll
        for (int i = 1; i < 4; ++i) {
            // reference: coef = sum_j conj(b_j)*v_i == conj(S)*v_i (linearity)
            const float cr = Sr * vr[i] + Si * vi[i];
            const float ci = Sr * vi[i] - Si * vr[i];
            // w = v_i - coef * S
            const float wr = vr[i] - (cr * Sr - ci * Si);
            const float wi = vi[i] - (cr * Si + ci * Sr);
            t   = wr + EPS;
            inv = __builtin_amdgcn_rsqf(t * t + wi * wi);
            br  = wr * inv; bi = wi * inv;
            o[2 * i][k] = br; o[2 * i + 1][k] = bi;
            Sr += br; Si += bi;
        }
    }

    #pragma unroll
    for (int c = 0; c < NCH; ++c) // streaming (non-temporal) b128 stores
        __builtin_nontemporal_store(*(v4f*)o[c],
                                    (v4f*)(out + base + (long)c * FT));
}

extern "C" void kernel_launch(void* const* d_in, const int* in_sizes, int n_in,
                              void* d_out, int out_size, void* d_ws, size_t ws_size,
                              hipStream_t stream) {
    const float* in  = (const float*)d_in[0]; // (B, 8, 257, 1000) f32
    float*       out = (float*)d_out;         // (B, 4, 2, 257, 1000) f32, same layout
    const int B = in_sizes[0] / (NCH * FT);   // 8
    const int threads = 256;                  // 8 wave32 waves per block
    dim3 grid((GPB + threads - 1) / threads, B); // 251 x 8 blocks, no int division
    gs_crm_kernel<<<grid, dim3(threads, 1, 1), 0, stream>>>(in, out);
}